// AttentionDecoder_47476568489991
// MI455X (gfx1250) — compile-verified
//
#include <hip/hip_runtime.h>

// Problem sizes (fixed by the reference)
#define BDIM 128
#define SDIM 256
#define EDIM 1024
#define VDIM 32000
#define E2   2048
#define E3   3072

// GEMM tiling
#define KC   128            // K chunk staged in LDS per pass
#define AP   132            // A LDS pitch (floats): 132 % 64 == 4 -> conflict-free b64 reads
#define BP   68             // B LDS pitch (floats) for non-transposed B
#define A_FLOATS (128 * AP)         // 16896 floats per A buffer
#define B_FLOATS (KC * BP)          // 8704 floats per B buffer (>= 64*AP=8448 for BT)
#define SMEM_FLOATS (2 * (A_FLOATS + B_FLOATS))   // 51200 floats = 204800 bytes

typedef __attribute__((ext_vector_type(2))) float v2f;
typedef __attribute__((ext_vector_type(8))) float v8f;

// Exact element type the async-LDS builtins expect: vector of 4 ints (16B)
typedef int v4i __attribute__((vector_size(4 * sizeof(int))));
typedef __attribute__((address_space(1))) v4i* as1_v4i;   // global
typedef __attribute__((address_space(3))) v4i* as3_v4i;   // LDS

// ---------------- gfx1250 async copy: global -> LDS, 16B per lane ----------------
// Pointers built via integer round-trip: generic LDS address low 32 bits == LDS offset.
__device__ __forceinline__ void async_ld16(float* ldst, const float* gsrc) {
#if __has_builtin(__builtin_amdgcn_global_load_async_to_lds_b128)
  as1_v4i g = (as1_v4i)(unsigned long long)gsrc;
  as3_v4i l = (as3_v4i)(unsigned)(unsigned long long)ldst;
  __builtin_amdgcn_global_load_async_to_lds_b128(g, l, 0, 0);
#else
  unsigned loff = (unsigned)(unsigned long long)ldst;  // LDS byte offset
  asm volatile("global_load_async_to_lds_b128 %0, %1, off"
               :: "v"(loff), "v"(gsrc) : "memory");
#endif
}
__device__ __forceinline__ void wait_async0() {
#if __has_builtin(__builtin_amdgcn_s_wait_asynccnt)
  __builtin_amdgcn_s_wait_asynccnt(0);
#else
  asm volatile("s_wait_asynccnt 0x0" ::: "memory");
#endif
}

// ---------------- wave32 / block reductions ----------------
__device__ __forceinline__ float warp_max(float v) {
#pragma unroll
  for (int o = 16; o > 0; o >>= 1) v = fmaxf(v, __shfl_xor(v, o, 32));
  return v;
}
__device__ __forceinline__ float warp_sum(float v) {
#pragma unroll
  for (int o = 16; o > 0; o >>= 1) v += __shfl_xor(v, o, 32);
  return v;
}
__device__ __forceinline__ float block_max(float v, float* sm) {
  v = warp_max(v);
  if ((threadIdx.x & 31) == 0) sm[threadIdx.x >> 5] = v;
  __syncthreads();
  float r = sm[0];
  const int nw = blockDim.x >> 5;
  for (int i = 1; i < nw; ++i) r = fmaxf(r, sm[i]);
  __syncthreads();
  return r;
}
__device__ __forceinline__ float block_sum(float v, float* sm) {
  v = warp_sum(v);
  if ((threadIdx.x & 31) == 0) sm[threadIdx.x >> 5] = v;
  __syncthreads();
  float r = sm[0];
  const int nw = blockDim.x >> 5;
  for (int i = 1; i < nw; ++i) r += sm[i];
  __syncthreads();
  return r;
}

// ---------------- fp32 WMMA GEMM with async double-buffered LDS staging ----------------
// C(128 x N) = A(128 x K) * B(K x N) + bias, one block per 64-column strip (reads B once).
// BT=false: Bm is (K x N) row-major, ldb = N   (B[k][n] = Bm[k*ldb + n])
// BT=true : Bm is (N x K) row-major, ldb = K   (B[k][n] = Bm[n*ldb + k])  -> C = A * W^T
// Block: 256 threads = 8 waves; wave w computes rows [16w,16w+16) x 64 cols
// (four 16x16 f32 WMMA accumulators). Requirements: N % 64 == 0, K % KC == 0.
template <bool BT>
__global__ __launch_bounds__(256) void gemm_wmma_f32(
    const float* __restrict__ A, int lda,
    const float* __restrict__ Bm, int ldb,
    const float* __restrict__ bias,
    float* __restrict__ C, int ldc, int K) {
  extern __shared__ float smem[];
  // buffer bases computed arithmetically (no LDS-pointer static initializers)

  const int t    = threadIdx.x;
  const int wave = t >> 5;
  const int lane = t & 31;
  const int lrow = lane & 15;   // A-frag row / B-frag column within 16
  const int half = lane >> 4;   // selects K pair {0,1} or {2,3}
  const int n0   = blockIdx.x * 64;
  const int mrow = wave * 16 + lrow;

  v8f acc[4];
#pragma unroll
  for (int i = 0; i < 4; ++i) acc[i] = (v8f){0.f,0.f,0.f,0.f,0.f,0.f,0.f,0.f};

  // ---- async stage of one K-chunk (issued by all 256 threads cooperatively) ----
  auto stage = [&](int buf, int k0) {
    float* Al = smem + buf * A_FLOATS;
    float* Bl = smem + 2 * A_FLOATS + buf * B_FLOATS;
    // A chunk: 128 rows x KC cols -> Al[row*AP + k], 16B lanes, fully coalesced
#pragma unroll
    for (int i = 0; i < (128 * (KC / 4)) / 256; ++i) {       // 16 iters
      const int idx = i * 256 + t;
      const int r  = idx >> 5;             // / (KC/4)
      const int s4 = (idx & 31) * 4;
      async_ld16(&Al[r * AP + s4], &A[(size_t)r * lda + k0 + s4]);
    }
    if (BT) {
      // W chunk: 64 n-rows x KC k-cols -> Bl[n*AP + k]
#pragma unroll
      for (int i = 0; i < (64 * (KC / 4)) / 256; ++i) {      // 8 iters
        const int idx = i * 256 + t;
        const int r  = idx >> 5;
        const int s4 = (idx & 31) * 4;
        async_ld16(&Bl[r * AP + s4], &Bm[(size_t)(n0 + r) * ldb + k0 + s4]);
      }
    } else {
      // B chunk: KC k-rows x 64 n-cols -> Bl[k*BP + n]
#pragma unroll
      for (int i = 0; i < (KC * 16) / 256; ++i) {            // 8 iters
        const int idx = i * 256 + t;
        const int r  = idx >> 4;
        const int s4 = (idx & 15) * 4;
        async_ld16(&Bl[r * BP + s4], &Bm[(size_t)(k0 + r) * ldb + n0 + s4]);
      }
    }
  };

  const int nc = K / KC;
  stage(0, 0);  // prologue

  for (int c = 0; c < nc; ++c) {
    wait_async0();        // this wave's async copies for chunk c are done
    __syncthreads();      // => all waves' copies are done; chunk c visible in LDS
    if (c + 1 < nc) stage((c + 1) & 1, (c + 1) * KC);  // DMA next chunk overlaps compute

    const float* Al = smem + (c & 1) * A_FLOATS;
    const float* Bl = smem + 2 * A_FLOATS + (c & 1) * B_FLOATS;
#pragma unroll 4
    for (int k = 0; k < KC; k += 4) {
      const v2f a = *(const v2f*)&Al[mrow * AP + k + 2 * half];
      v2f b0, b1, b2, b3;
      if (BT) {
        b0 = *(const v2f*)&Bl[(lrow +  0) * AP + k + 2 * half];
        b1 = *(const v2f*)&Bl[(lrow + 16) * AP + k + 2 * half];
        b2 = *(const v2f*)&Bl[(lrow + 32) * AP + k + 2 * half];
        b3 = *(const v2f*)&Bl[(lrow + 48) * AP + k + 2 * half];
      } else {
        const int kg = k + 2 * half;
        const float* r0 = &Bl[kg * BP + lrow];
        const float* r1 = &Bl[(kg + 1) * BP + lrow];
        b0.x = r0[0];  b0.y = r1[0];
        b1.x = r0[16]; b1.y = r1[16];
        b2.x = r0[32]; b2.y = r1[32];
        b3.x = r0[48]; b3.y = r1[48];
      }
      acc[0] = __builtin_amdgcn_wmma_f32_16x16x4_f32(false, a, false, b0, (short)0, acc[0], false, false);
      acc[1] = __builtin_amdgcn_wmma_f32_16x16x4_f32(false, a, false, b1, (short)0, acc[1], false, false);
      acc[2] = __builtin_amdgcn_wmma_f32_16x16x4_f32(false, a, false, b2, (short)0, acc[2], false, false);
      acc[3] = __builtin_amdgcn_wmma_f32_16x16x4_f32(false, a, false, b3, (short)0, acc[3], false, false);
    }
    __syncthreads();  // all waves done with buffer (c&1) before its next DMA overwrite
  }

  // C/D layout: VGPR r, lanes<16 -> row r, lanes>=16 -> row r+8; col = lane&15
#pragma unroll
  for (int i = 0; i < 4; ++i) {
    const int col = n0 + i * 16 + lrow;
    const float bv = bias ? bias[col] : 0.0f;
#pragma unroll
    for (int r = 0; r < 8; ++r)
      C[(size_t)(wave * 16 + 8 * half + r) * ldc + col] = acc[i][r] + bv;
  }
}

// ---------------- small elementwise / reduction kernels ----------------

// combined[b] = [emb[idx[b]], h0[b]] ; also cat2[b][E:2E] = x0[b]
__global__ __launch_bounds__(256) void k_embed(
    const int* __restrict__ idx, const float* __restrict__ emb,
    const float* __restrict__ h0, float* __restrict__ combined,
    float* __restrict__ cat2) {
  const int b = blockIdx.x;
  const int row = idx[b];
  for (int e = threadIdx.x; e < EDIM; e += 256) {
    const float xv = emb[(size_t)row * EDIM + e];
    const float hv = h0[(size_t)b * EDIM + e];
    combined[(size_t)b * E2 + e] = xv;
    combined[(size_t)b * E2 + EDIM + e] = hv;
    cat2[(size_t)b * E2 + EDIM + e] = xv;
  }
}

// softmax over S=256 per row; write to workspace and to output slot 3
__global__ __launch_bounds__(256) void k_softmax(
    const float* __restrict__ scores, float* __restrict__ w_ws,
    float* __restrict__ w_out) {
  __shared__ float sm[8];
  const int b = blockIdx.x, t = threadIdx.x;
  const float x = scores[(size_t)b * SDIM + t];
  const float mx = block_max(x, sm);
  const float e = __expf(x - mx);
  const float s = block_sum(e, sm);
  const float w = e / s;
  w_ws[(size_t)b * SDIM + t] = w;
  w_out[(size_t)b * SDIM + t] = w;
}

// cat2[b][0:E] = sum_s w[b,s] * enc[b,s,:]   (pure HBM stream over encoder_output)
__global__ __launch_bounds__(256) void k_attn_apply(
    const float* __restrict__ w, const float* __restrict__ enc,
    float* __restrict__ cat2) {
  __shared__ float ws[SDIM];
  const int b = blockIdx.y;
  const int e = blockIdx.x * 256 + threadIdx.x;
  ws[threadIdx.x] = w[(size_t)b * SDIM + threadIdx.x];
  __syncthreads();
  float acc = 0.f;
  const float* ep = &enc[(size_t)b * SDIM * EDIM + e];
#pragma unroll 4
  for (int s = 0; s < SDIM; ++s) acc = fmaf(ws[s], ep[(size_t)s * EDIM], acc);
  cat2[(size_t)b * E2 + e] = acc;
}

// GRU gates (biases already folded into gi/gh by the GEMMs)
__global__ __launch_bounds__(256) void k_gates(
    const float* __restrict__ gi, const float* __restrict__ gh,
    const float* __restrict__ h0, float* __restrict__ hnew_ws,
    float* __restrict__ hnew_out) {
  const int b = blockIdx.y;
  const int e = blockIdx.x * 256 + threadIdx.x;
  const float ir  = gi[(size_t)b * E3 + e];
  const float iz  = gi[(size_t)b * E3 + EDIM + e];
  const float in_ = gi[(size_t)b * E3 + 2 * EDIM + e];
  const float hr  = gh[(size_t)b * E3 + e];
  const float hz  = gh[(size_t)b * E3 + EDIM + e];
  const float hn  = gh[(size_t)b * E3 + 2 * EDIM + e];
  const float r = 1.f / (1.f + __expf(-(ir + hr)));
  const float z = 1.f / (1.f + __expf(-(iz + hz)));
  const float n = tanhf(in_ + r * hn);
  const float h = h0[(size_t)b * EDIM + e];
  const float o = (1.f - z) * n + z * h;
  hnew_ws[(size_t)b * EDIM + e] = o;
  hnew_out[(size_t)b * EDIM + e] = o;
}

// log_softmax over V=32000 per row (logits live in L2-resident workspace)
__global__ __launch_bounds__(256) void k_logsoftmax(
    const float* __restrict__ logits, float* __restrict__ out) {
  __shared__ float sm[8];
  const int b = blockIdx.x, t = threadIdx.x;
  const float* row = &logits[(size_t)b * VDIM];
  float mx = -__builtin_huge_valf();
  for (int i = t; i < VDIM; i += 256) mx = fmaxf(mx, row[i]);
  mx = block_max(mx, sm);
  float s = 0.f;
  for (int i = t; i < VDIM; i += 256) s += __expf(row[i] - mx);
  s = block_sum(s, sm);
  const float lse = __logf(s) + mx;
  float* orow = &out[(size_t)b * VDIM];
  for (int i = t; i < VDIM; i += 256) orow[i] = row[i] - lse;
}

// ---------------- launch ----------------
extern "C" void kernel_launch(void* const* d_in, const int* in_sizes, int n_in,
                              void* d_out, int out_size, void* d_ws, size_t ws_size,
                              hipStream_t stream) {
  const int*   idx   = (const int*)d_in[0];
  const float* h0    = (const float*)d_in[1];   // (1,B,E) == (B,E) flat
  const float* enc   = (const float*)d_in[2];
  const float* emb   = (const float*)d_in[3];
  const float* attnW = (const float*)d_in[4];
  const float* attnB = (const float*)d_in[5];
  const float* combW = (const float*)d_in[6];
  const float* combB = (const float*)d_in[7];
  const float* Wih   = (const float*)d_in[8];
  const float* Whh   = (const float*)d_in[9];
  const float* bih   = (const float*)d_in[10];
  const float* bhh   = (const float*)d_in[11];
  const float* outW  = (const float*)d_in[12];
  const float* outB  = (const float*)d_in[13];

  float* out      = (float*)d_out;
  float* out_logp = out;                                   // B*V
  float* out_h    = out + (size_t)BDIM * VDIM;             // B*E
  float* out_attn = out_h + (size_t)BDIM * EDIM;           // B*S

  // workspace layout (floats); total ~5.73M floats (~22.9 MB)
  float* ws = (float*)d_ws;
  size_t o = 0;
  float* combined = ws + o; o += (size_t)BDIM * E2;
  float* scores   = ws + o; o += (size_t)BDIM * SDIM;
  float* attw     = ws + o; o += (size_t)BDIM * SDIM;
  float* cat2     = ws + o; o += (size_t)BDIM * E2;
  float* gruin    = ws + o; o += (size_t)BDIM * EDIM;
  float* gi       = ws + o; o += (size_t)BDIM * E3;
  float* gh       = ws + o; o += (size_t)BDIM * E3;
  float* hnew     = ws + o; o += (size_t)BDIM * EDIM;
  float* logits   = ws + o; o += (size_t)BDIM * VDIM;

  const size_t smem_bytes = SMEM_FLOATS * sizeof(float);  // 204800 B (<320KB/WGP)

  // 1) gather embedding + build concat buffers
  k_embed<<<BDIM, 256, 0, stream>>>(idx, emb, h0, combined, cat2);

  // 2) attention scores: (128x2048) @ (2048x256) + attn_b
  gemm_wmma_f32<false><<<SDIM / 64, 256, smem_bytes, stream>>>(
      combined, E2, attnW, SDIM, attnB, scores, SDIM, E2);

  // 3) softmax over S
  k_softmax<<<BDIM, SDIM, 0, stream>>>(scores, attw, out_attn);

  // 4) attn_applied: bandwidth-bound stream of encoder_output (134 MB)
  k_attn_apply<<<dim3(EDIM / 256, BDIM), 256, 0, stream>>>(attw, enc, cat2);

  // 5) gru_in: (128x2048) @ (2048x1024) + comb_b
  gemm_wmma_f32<false><<<EDIM / 64, 256, smem_bytes, stream>>>(
      cat2, E2, combW, EDIM, combB, gruin, EDIM, E2);

  // 6) gi = gru_in @ W_ih^T + b_ih ; gh = h0 @ W_hh^T + b_hh
  gemm_wmma_f32<true><<<E3 / 64, 256, smem_bytes, stream>>>(
      gruin, EDIM, Wih, EDIM, bih, gi, E3, EDIM);
  gemm_wmma_f32<true><<<E3 / 64, 256, smem_bytes, stream>>>(
      h0, EDIM, Whh, EDIM, bhh, gh, E3, EDIM);

  // 7) GRU gates -> h_new (also output 2)
  k_gates<<<dim3(EDIM / 256, BDIM), 256, 0, stream>>>(gi, gh, h0, hnew, out_h);

  // 8) logits: (128x1024) @ (1024x32000) + out_b — weights streamed exactly once
  gemm_wmma_f32<false><<<VDIM / 64, 256, smem_bytes, stream>>>(
      hnew, EDIM, outW, VDIM, outB, logits, VDIM, EDIM);

  // 9) log_softmax over V -> output 1
  k_logsoftmax<<<BDIM, 256, 0, stream>>>(logits, out_logp);
}